// GeneralEdgeConv_4363686772851
// MI455X (gfx1250) — compile-verified
//
#include <hip/hip_runtime.h>
#include <hip/hip_bf16.h>

typedef float v2f __attribute__((ext_vector_type(2)));
typedef float v8f __attribute__((ext_vector_type(8)));
typedef unsigned int u32x4 __attribute__((ext_vector_type(4)));
typedef int i32x8 __attribute__((ext_vector_type(8)));
typedef int i32x4 __attribute__((ext_vector_type(4)));

#define LDS_PITCH 132  // 128 + 4 floats pad: stride 132 -> 16 distinct banks across A-rows
#define MT_N 32        // rows per block, node GEMM (3 fused chains -> keep acc regs bounded)
#define MT_E 64        // edges per block, edge GEMM (4 M-tiles share each B fragment)

// ---------------------------------------------------------------------------
// TDM: DMA a 2D tile (nrows x 128 f32) from global to LDS, inserting 4 dwords
// of padding after every 128 dwords (-> LDS_PITCH row stride).
// D# layout per CDNA5 ISA. rows_avail = rows
// remaining in the tensor; hardware zero-fills reads beyond it.
// clang-23 lane -> 6-arg builtin (g0,g1,g2,g3,extra,cpol).
// ---------------------------------------------------------------------------
__device__ __forceinline__ void tdm_load_tile_2d(const void* gptr,
                                                 unsigned lds_byte_off,
                                                 int rows_avail, int nrows) {
    unsigned long long ga = (unsigned long long)(uintptr_t)gptr;
    u32x4 g0;
    g0[0] = 1u;                                    // count=1 (valid user descriptor)
    g0[1] = lds_byte_off;                          // lds_addr (bytes)
    g0[2] = (unsigned)ga;                          // global_addr[31:0]
    g0[3] = (unsigned)((ga >> 32) & 0x1FFFFFFu)    // global_addr[56:32]
          | (2u << 30);                            // type = 2 ("image")
    i32x8 g1;
    g1[0] = (2 << 16)      // data_size = 4 bytes
          | (1 << 20)      // pad_enable
          | (6 << 22)      // pad_interval: code 6 -> every 128 dwords
          | (3 << 25);     // pad_amount:   code 3 -> 4 dwords
    g1[1] = (128 << 16);                           // tensor_dim0 = 128 (bits 79:48)
    g1[2] = (rows_avail & 0xFFFF) << 16;           // tensor_dim1[15:0] (bits 111:80)
    g1[3] = ((rows_avail >> 16) & 0xFFFF)          // tensor_dim1[31:16]
          | (128 << 16);                           // tile_dim0 = 128 (bits 127:112)
    g1[4] = nrows;                                 // tile_dim1, tile_dim2 = 0
    g1[5] = 128;                                   // tensor_dim0_stride (bits 207:160)
    g1[6] = 0;
    g1[7] = 0;
    i32x4 gz4 = {0, 0, 0, 0};                      // groups 2/3 unused (2D tensor)
    i32x8 gz8 = {0, 0, 0, 0, 0, 0, 0, 0};
    __builtin_amdgcn_tensor_load_to_lds(g0, g1, gz4, gz4, gz8, 0);
}

// ---------------------------------------------------------------------------
// wsum[n] = sum of edge_weight over edges with dst == n  (weighted in-degree)
// ---------------------------------------------------------------------------
__global__ __launch_bounds__(256) void zero_f32(float* __restrict__ p, int n) {
    int i = blockIdx.x * 256 + threadIdx.x;
    if (i < n) p[i] = 0.0f;
}

__global__ __launch_bounds__(256) void wsum_accum(
    const int* __restrict__ dst, const float* __restrict__ ew,
    float* __restrict__ wsum, int E) {
    int e = blockIdx.x * 256 + threadIdx.x;
    if (e < E) unsafeAtomicAdd(&wsum[dst[e]], ew[e]);
}

// ---------------------------------------------------------------------------
// Pack weight rows pairwise: P[(k/2)*128 + col] = (W[k][col], W[k+1][col]).
// Turns each WMMA B-fragment into a single global_load_b64.
// ---------------------------------------------------------------------------
__global__ __launch_bounds__(256) void pack_weights(
    const float* __restrict__ W, v2f* __restrict__ P, int nrowpairs) {
    int i = blockIdx.x * 256 + threadIdx.x;
    if (i < nrowpairs * 128) {
        int k2 = i >> 7, col = i & 127;
        v2f v;
        v.x = W[(size_t)(2 * k2)     * 128 + col];
        v.y = W[(size_t)(2 * k2 + 1) * 128 + col];
        P[i] = v;
    }
}

// ---------------------------------------------------------------------------
// Kernel 1: 32-row stripe of x -> three WMMA GEMM chains (W_i, W_j, W_self).
// Epilogue fuses: out = x@W_self + wsum * (x@W_i);  h_j = x@W_j to workspace.
// ---------------------------------------------------------------------------
__global__ __launch_bounds__(256) void edgeconv_node_gemm(
    const float* __restrict__ x,
    const v2f* __restrict__ P_i,
    const v2f* __restrict__ P_j,
    const v2f* __restrict__ P_s,
    const float* __restrict__ wsum,
    float* __restrict__ h_j,
    float* __restrict__ out,
    int N)
{
    __shared__ float lds_x[MT_N * LDS_PITCH];
    __shared__ float s_ws[MT_N];

    const int row0 = blockIdx.x * MT_N;
    const int t    = threadIdx.x;
    const bool full = (row0 + MT_N) <= N;

    if (t < 32) {  // wave 0 issues the DMA; TENSORcnt is per-wave
        tdm_load_tile_2d(x + (size_t)row0 * 128,
                         (unsigned)(uintptr_t)&lds_x[0], N - row0, MT_N);
        int gr = row0 + t; if (gr >= N) gr = N - 1;
        s_ws[t] = wsum[gr];
        __builtin_amdgcn_s_wait_tensorcnt(0);
    }
    __syncthreads();

    const int wave  = t >> 5;
    const int lane  = t & 31;
    const int lrow  = lane & 15;
    const int khalf = (lane >> 4) * 2;    // lanes 16-31 hold K=k+2,k+3 (ISA 7.12.2)
    const int col   = wave * 16 + lrow;

    const v2f* Pw[3] = { P_i, P_j, P_s };
    v8f acc[3][2] = {};

    #pragma unroll 4
    for (int k = 0; k < 128; k += 4) {
        v2f a0, a1;
        a0.x = lds_x[lrow * LDS_PITCH + k + khalf];
        a0.y = lds_x[lrow * LDS_PITCH + k + khalf + 1];
        a1.x = lds_x[(lrow + 16) * LDS_PITCH + k + khalf];
        a1.y = lds_x[(lrow + 16) * LDS_PITCH + k + khalf + 1];
        const size_t bidx = (size_t)((k + khalf) >> 1) * 128 + col;
        #pragma unroll
        for (int p = 0; p < 3; ++p) {
            v2f b = Pw[p][bidx];                          // one global_load_b64
            acc[p][0] = __builtin_amdgcn_wmma_f32_16x16x4_f32(
                false, a0, false, b, (short)0, acc[p][0], false, false);
            acc[p][1] = __builtin_amdgcn_wmma_f32_16x16x4_f32(
                false, a1, false, b, (short)0, acc[p][1], false, false);
        }
    }

    const int mbase = (lane >> 4) << 3;   // C/D: VGPR r <-> M = r (+8 for upper lanes)
    if (full) {
        #pragma unroll
        for (int mt = 0; mt < 2; ++mt) {
            #pragma unroll
            for (int r = 0; r < 8; ++r) {
                int m = mt * 16 + r + mbase;
                size_t off = (size_t)(row0 + m) * 128 + col;
                h_j[off] = acc[1][mt][r];
                out[off] = acc[2][mt][r] + s_ws[m] * acc[0][mt][r];
            }
        }
    } else {
        #pragma unroll
        for (int mt = 0; mt < 2; ++mt) {
            #pragma unroll
            for (int r = 0; r < 8; ++r) {
                int m = mt * 16 + r + mbase;
                int row = row0 + m;
                if (row < N) {
                    size_t off = (size_t)row * 128 + col;
                    h_j[off] = acc[1][mt][r];
                    out[off] = acc[2][mt][r] + s_ws[m] * acc[0][mt][r];
                }
            }
        }
    }
}

// ---------------------------------------------------------------------------
// Kernel 2: per 64-edge tile: val = ew * (edge_attr@W_e + h_j[src]),
// atomically accumulated into out[dst]. Four 16-edge M-tiles share each B
// fragment (4 WMMA per global_load_b64). h_j/out (~102 MB) are L2-resident.
// ---------------------------------------------------------------------------
__global__ __launch_bounds__(256) void edgeconv_edge_kernel(
    const float* __restrict__ edge_attr,
    const int*   __restrict__ src,
    const int*   __restrict__ dst,
    const float* __restrict__ ew,
    const v2f*   __restrict__ P_e,
    const float* __restrict__ h_j,
    float* __restrict__ out,
    int E)
{
    __shared__ float lds_a[MT_E * LDS_PITCH];
    __shared__ int   s_src[MT_E];
    __shared__ int   s_dst[MT_E];
    __shared__ float s_w[MT_E];

    const int e0 = blockIdx.x * MT_E;
    const int t  = threadIdx.x;
    const bool full = (e0 + MT_E) <= E;

    if (t < 32) {
        tdm_load_tile_2d(edge_attr + (size_t)e0 * 128,
                         (unsigned)(uintptr_t)&lds_a[0], E - e0, MT_E);
        #pragma unroll
        for (int i = 0; i < 2; ++i) {
            int m = t + i * 32;
            int ge = e0 + m; if (ge >= E) ge = E - 1;
            s_src[m] = src[ge];
            s_dst[m] = dst[ge];
            s_w[m]   = ew[ge];
        }
        __builtin_amdgcn_s_wait_tensorcnt(0);
    }
    __syncthreads();

    const int wave  = t >> 5;
    const int lane  = t & 31;
    const int lrow  = lane & 15;
    const int khalf = (lane >> 4) * 2;
    const int col   = wave * 16 + lrow;

    v8f acc[4] = {};
    #pragma unroll 4
    for (int k = 0; k < 128; k += 4) {
        v2f b = P_e[(size_t)((k + khalf) >> 1) * 128 + col];
        #pragma unroll
        for (int mt = 0; mt < 4; ++mt) {
            v2f a;
            a.x = lds_a[(lrow + mt * 16) * LDS_PITCH + k + khalf];
            a.y = lds_a[(lrow + mt * 16) * LDS_PITCH + k + khalf + 1];
            acc[mt] = __builtin_amdgcn_wmma_f32_16x16x4_f32(
                false, a, false, b, (short)0, acc[mt], false, false);
        }
    }

    const int mbase = (lane >> 4) << 3;
    if (full) {
        #pragma unroll
        for (int mt = 0; mt < 4; ++mt) {
            #pragma unroll
            for (int r = 0; r < 8; ++r) {
                int m = mt * 16 + r + mbase;
                int   di = s_dst[m];
                int   si = s_src[m];
                float val = s_w[m] * (acc[mt][r] + h_j[(size_t)si * 128 + col]);
                unsafeAtomicAdd(&out[(size_t)di * 128 + col], val);
            }
        }
    } else {
        #pragma unroll
        for (int mt = 0; mt < 4; ++mt) {
            #pragma unroll
            for (int r = 0; r < 8; ++r) {
                int m = mt * 16 + r + mbase;
                if (e0 + m < E) {
                    int   di = s_dst[m];
                    int   si = s_src[m];
                    float val = s_w[m] * (acc[mt][r] + h_j[(size_t)si * 128 + col]);
                    unsafeAtomicAdd(&out[(size_t)di * 128 + col], val);
                }
            }
        }
    }
}

// ---------------------------------------------------------------------------
// Launch. Inputs: 0:x (N*128) 1:edge_index (2E i32) 2:edge_weight (E)
//                 3:edge_attr (E*128) 4:W_msg (384*128) 5:W_self (128*128)
// Workspace: h_j (N*128 f32) | wsum (N f32) | P_msg (384*128 f32) | P_self (128*128 f32)
// ---------------------------------------------------------------------------
extern "C" void kernel_launch(void* const* d_in, const int* in_sizes, int n_in,
                              void* d_out, int out_size, void* d_ws, size_t ws_size,
                              hipStream_t stream) {
    const float* x      = (const float*)d_in[0];
    const int*   eidx   = (const int*)  d_in[1];
    const float* ew     = (const float*)d_in[2];
    const float* eattr  = (const float*)d_in[3];
    const float* W_msg  = (const float*)d_in[4];
    const float* W_self = (const float*)d_in[5];
    float*       out    = (float*)d_out;

    const int N = in_sizes[0] / 128;
    const int E = in_sizes[2];
    const int* src = eidx;        // edge_index[0]
    const int* dst = eidx + E;    // edge_index[1]

    float* h_j     = (float*)d_ws;                   // N*128
    float* wsum    = h_j + (size_t)N * 128;          // N
    float* P_msgf  = wsum + N;                       // 384*128 packed
    float* P_selff = P_msgf + 384 * 128;             // 128*128 packed
    v2f* P_msg  = (v2f*)P_msgf;
    v2f* P_self = (v2f*)P_selff;
    const v2f* P_i = P_msg;                          // packed W_i  (64 row-pairs)
    const v2f* P_j = P_msg + 64 * 128;               // packed W_j
    const v2f* P_e = P_msg + 128 * 128;              // packed W_e
    const v2f* P_s = P_self;                         // packed W_self

    dim3 blk(256);

    zero_f32<<<(N + 255) / 256, blk, 0, stream>>>(wsum, N);
    wsum_accum<<<(E + 255) / 256, blk, 0, stream>>>(dst, ew, wsum, E);
    pack_weights<<<(384 / 2 * 128 + 255) / 256, blk, 0, stream>>>(W_msg, P_msg, 384 / 2);
    pack_weights<<<(128 / 2 * 128 + 255) / 256, blk, 0, stream>>>(W_self, P_self, 128 / 2);

    edgeconv_node_gemm<<<(N + MT_N - 1) / MT_N, blk, 0, stream>>>(
        x, P_i, P_j, P_s, wsum, h_j, out, N);

    edgeconv_edge_kernel<<<(E + MT_E - 1) / MT_E, blk, 0, stream>>>(
        eattr, src, dst, ew, P_e, h_j, out, E);
}